// NMSLayer_46497315946961
// MI455X (gfx1250) — compile-verified
//
#include <hip/hip_runtime.h>
#include <hip/hip_bf16.h>
#include <stdint.h>

// ---------------- problem constants (from reference) ----------------
#define BB   2          // batch
#define NN   10647      // boxes per image
#define CC   85         // 5 + 80 channels
#define NCLS 80
#define MM   16384      // next pow2 >= NN (bitonic sort size)
#define BT   1024       // threads for the per-image kernels
#define CONF_TH 0.5f
#define NMS_TH  0.5f

// ---------------- CDNA5 feature detection ----------------
#define __AS1 __attribute__((address_space(1)))
#define __AS3 __attribute__((address_space(3)))

typedef __attribute__((__vector_size__(4 * sizeof(int)))) int v4i;

#if defined(__has_builtin)
#  if __has_builtin(__builtin_amdgcn_global_load_async_to_lds_b128)
#    define HAVE_ASYNC_B128 1
#  endif
#  if __has_builtin(__builtin_amdgcn_s_wait_asynccnt)
#    define HAVE_WAIT_ASYNC 1
#  endif
#endif

__device__ __forceinline__ void wait_asynccnt0() {
#if defined(HAVE_WAIT_ASYNC)
    __builtin_amdgcn_s_wait_asynccnt(0);
#else
    asm volatile("s_wait_asynccnt 0" ::: "memory");
#endif
}

// ---------------- workspace layout (bytes) ----------------
// boxes  : BB*NN * float4
// conf   : BB*NN * float
// score  : BB*NN * float
// clsf   : BB*NN * float
// keys   : BB*MM * u64
// keep   : BB*NN * u32
#define OFF_BOXES 0
#define SZ_BOXES  ((size_t)BB * NN * 16)
#define OFF_CONF  (OFF_BOXES + SZ_BOXES)
#define SZ_CONF   ((size_t)BB * NN * 4)
#define OFF_SCORE (OFF_CONF + SZ_CONF)
#define SZ_SCORE  ((size_t)BB * NN * 4)
#define OFF_CLS   (OFF_SCORE + SZ_SCORE)
#define SZ_CLS    ((size_t)BB * NN * 4)
#define OFF_KEYS  (OFF_CLS + SZ_CLS)
#define SZ_KEYS   ((size_t)BB * MM * 8)
#define OFF_KEEP  (OFF_KEYS + SZ_KEYS)

// LDS layout for the NMS kernel
#define SBOX_BYTES  ((size_t)NN * 16)            // 170352
#define SKEEP_WORDS (MM / 32)                    // 512
#define SKEEP_BYTES ((size_t)SKEEP_WORDS * 4)    // 2048
#define SCLS_BYTES  ((size_t)NN)                 // 10647
#define SMEM_NMS    ((SBOX_BYTES + SKEEP_BYTES + SCLS_BYTES + 15) & ~(size_t)15)

// ================= K1: preprocess =================
// mask by conf, center->corner, class max/argmax, build stable sort key
__global__ void k_prep(const float* __restrict__ x,
                       float4* __restrict__ boxes,
                       float*  __restrict__ conf,
                       float*  __restrict__ score,
                       float*  __restrict__ clsf,
                       unsigned long long* __restrict__ keys) {
    int t = blockIdx.x * blockDim.x + threadIdx.x;
    int b = t / MM;
    int i = t - b * MM;
    if (b >= BB) return;
    if (i >= NN) { keys[(size_t)b * MM + i] = 0ull; return; }  // padding sorts last

    const float* p = x + ((size_t)b * NN + i) * CC;
    __builtin_prefetch(p, 0, 1);           // global_prefetch_b8 on gfx1250
    __builtin_prefetch(p + 64, 0, 1);

    float cr = p[4];
    bool  valid = cr > CONF_TH;
    float m = valid ? 1.0f : 0.0f;
    float cx = p[0] * m, cy = p[1] * m, w = p[2] * m, h = p[3] * m;
    float4 bx = make_float4(cx - 0.5f * w, cy - 0.5f * h,
                            cx + 0.5f * w, cy + 0.5f * h);
    float cf = cr * m;

    float best = p[5] * m;   // masked scores; argmax keeps first max (strict >)
    int   bc = 0;
    #pragma unroll 4
    for (int k = 1; k < NCLS; ++k) {
        float v = p[5 + k] * m;
        if (v > best) { best = v; bc = k; }
    }

    size_t o = (size_t)b * NN + i;
    boxes[o] = bx;
    conf[o]  = cf;
    score[o] = best;
    clsf[o]  = (float)bc;

    // key = conf bits (non-negative float => bit order == value order)
    //       || (0xFFFFFFFF - idx) so ties break by ascending index under a
    //       descending sort (matches JAX's stable argsort(-conf)).
    unsigned cb = __float_as_uint(cf);
    keys[(size_t)b * MM + i] =
        ((unsigned long long)cb << 32) | (0xFFFFFFFFu - (unsigned)i);
}

// ================= K2: in-LDS bitonic sort (descending), one WG per image ===
__global__ __launch_bounds__(BT)
void k_sort(unsigned long long* __restrict__ keys) {
    extern __shared__ unsigned long long skey[];   // MM * 8 = 128 KB in LDS
    const int b = blockIdx.x;
    const int tid = threadIdx.x;
    unsigned long long* gk = keys + (size_t)b * MM;

    for (int i = tid; i < MM; i += BT) skey[i] = gk[i];
    __syncthreads();

    for (unsigned k = 2; k <= MM; k <<= 1) {
        for (unsigned j = k >> 1; j > 0; j >>= 1) {
            for (unsigned i = tid; i < MM; i += BT) {
                unsigned ixj = i ^ j;
                if (ixj > i) {
                    bool up = ((i & k) == 0);     // "up" block => larger-first
                    unsigned long long a = skey[i];
                    unsigned long long c = skey[ixj];
                    bool sw = up ? (a < c) : (a > c);
                    if (sw) { skey[i] = c; skey[ixj] = a; }
                }
            }
            __syncthreads();
        }
    }

    for (int i = tid; i < MM; i += BT) gk[i] = skey[i];
}

// ================= K3: greedy NMS, one WG per image, all state in LDS =======
__global__ __launch_bounds__(BT)
void k_nms(const float4* __restrict__ boxes,
           const float*  __restrict__ clsf,
           const unsigned long long* __restrict__ keys,
           unsigned* __restrict__ keepflag) {
    extern __shared__ char sm[];
    float4*        sbox  = (float4*)sm;
    unsigned*      skeep = (unsigned*)(sm + SBOX_BYTES);
    unsigned char* scls  = (unsigned char*)(sm + SBOX_BYTES + SKEEP_BYTES);

    const int b = blockIdx.x;
    const int tid = threadIdx.x;
    const unsigned long long* gk = keys + (size_t)b * MM;

    for (int i = tid; i < NN; i += BT) keepflag[(size_t)b * NN + i] = 0u;
    for (int w = tid; w < SKEEP_WORDS; w += BT) skeep[w] = 0u;
    __syncthreads();

    // Gather sorted boxes/classes into LDS (per-lane scattered addresses,
    // async global->LDS path on gfx1250, tracked by ASYNCcnt).
    for (int r = tid; r < NN; r += BT) {
        unsigned long long key = gk[r];
        unsigned orig = 0xFFFFFFFFu - (unsigned)(key & 0xFFFFFFFFull);
        const float4* gp = boxes + (size_t)b * NN + orig;
#if defined(HAVE_ASYNC_B128)
        __builtin_amdgcn_global_load_async_to_lds_b128(
            (__AS1 v4i*)(const void*)gp,
            (__AS3 v4i*)(void*)&sbox[r], 0, 0);
#else
        sbox[r] = *gp;
#endif
        scls[r] = (unsigned char)clsf[(size_t)b * NN + orig];
        float cf = __uint_as_float((unsigned)(key >> 32));
        if (cf > CONF_TH) atomicOr(&skeep[r >> 5], 1u << (r & 31));
    }
    wait_asynccnt0();
    __syncthreads();

    // Greedy sequential suppression (matches the reference fori_loop exactly:
    // clears for j>i are independent of keep[j], so atomicAnd races are
    // idempotent and keep[i] is never touched during iteration i).
    for (int i = 0; i < NN; ++i) {
        if (!((skeep[i >> 5] >> (i & 31)) & 1u)) continue;   // uniform branch
        float4 bi = sbox[i];
        int    ci = scls[i];
        float  ai = (bi.z - bi.x) * (bi.w - bi.y);
        for (int j = i + 1 + tid; j < NN; j += BT) {
            if (!((skeep[j >> 5] >> (j & 31)) & 1u)) continue;
            if ((int)scls[j] != ci) continue;
            float4 bj = sbox[j];
            float ix1 = fmaxf(bi.x, bj.x), iy1 = fmaxf(bi.y, bj.y);
            float ix2 = fminf(bi.z, bj.z), iy2 = fminf(bi.w, bj.w);
            float iw = fmaxf(ix2 - ix1, 0.0f), ih = fmaxf(iy2 - iy1, 0.0f);
            float inter = iw * ih;
            float aj = (bj.z - bj.x) * (bj.w - bj.y);
            float uni = fmaxf(ai + aj - inter, 1e-9f);
            if (inter >= NMS_TH * uni)
                atomicAnd(&skeep[j >> 5], ~(1u << (j & 31)));
        }
        __syncthreads();
    }

    // Scatter keep bits back to original indices.
    for (int r = tid; r < NN; r += BT) {
        if ((skeep[r >> 5] >> (r & 31)) & 1u) {
            unsigned long long key = gk[r];
            unsigned orig = 0xFFFFFFFFu - (unsigned)(key & 0xFFFFFFFFull);
            keepflag[(size_t)b * NN + orig] = 1u;
        }
    }
}

// ================= K4: emit (B, N, 8) output =================
__global__ void k_out(const float4* __restrict__ boxes,
                      const float*  __restrict__ conf,
                      const float*  __restrict__ score,
                      const float*  __restrict__ clsf,
                      const unsigned* __restrict__ keepflag,
                      float* __restrict__ out) {
    int t = blockIdx.x * blockDim.x + threadIdx.x;
    if (t >= BB * NN) return;
    int b = t / NN;
    float km = keepflag[t] ? 1.0f : 0.0f;
    float4 bx = boxes[t];
    float* o = out + (size_t)t * 8;
    o[0] = (float)b * km;
    o[1] = bx.x * km;
    o[2] = bx.y * km;
    o[3] = bx.z * km;
    o[4] = bx.w * km;
    o[5] = conf[t]  * km;
    o[6] = score[t] * km;
    o[7] = clsf[t]  * km;
}

// ================= host launcher =================
extern "C" void kernel_launch(void* const* d_in, const int* in_sizes, int n_in,
                              void* d_out, int out_size, void* d_ws, size_t ws_size,
                              hipStream_t stream) {
    (void)in_sizes; (void)n_in; (void)out_size; (void)ws_size;
    const float* x = (const float*)d_in[0];
    char* ws = (char*)d_ws;

    float4*             boxes = (float4*)(ws + OFF_BOXES);
    float*              conf  = (float*)(ws + OFF_CONF);
    float*              score = (float*)(ws + OFF_SCORE);
    float*              clsf  = (float*)(ws + OFF_CLS);
    unsigned long long* keys  = (unsigned long long*)(ws + OFF_KEYS);
    unsigned*           keep  = (unsigned*)(ws + OFF_KEEP);
    float*              out   = (float*)d_out;

    {   // K1: preprocess, covers BB*MM threads (padding writes key=0)
        int total = BB * MM;
        k_prep<<<(total + 255) / 256, 256, 0, stream>>>(x, boxes, conf, score, clsf, keys);
    }
    // K2: bitonic sort of MM 64-bit keys per image, fully in LDS (128 KB)
    k_sort<<<BB, BT, (size_t)MM * 8, stream>>>(keys);
    // K3: greedy NMS with the whole sorted working set in one WGP's LDS (~183 KB)
    k_nms<<<BB, BT, SMEM_NMS, stream>>>(boxes, clsf, keys, keep);
    {   // K4: output
        int total = BB * NN;
        k_out<<<(total + 255) / 256, 256, 0, stream>>>(boxes, conf, score, clsf, keep, out);
    }
}